// BetaTCVAEModKLDLoss_61727269978467
// MI455X (gfx1250) — compile-verified
//
#include <hip/hip_runtime.h>
#include <math.h>

typedef float v2f __attribute__((ext_vector_type(2)));
typedef float v8f __attribute__((ext_vector_type(8)));
typedef unsigned int u32x4 __attribute__((ext_vector_type(4)));
typedef int i32x4 __attribute__((ext_vector_type(4)));
typedef int i32x8 __attribute__((ext_vector_type(8)));

#define LOG2PI_F 1.8378770664093453f

__device__ __forceinline__ float wave_sum(float v) {
#pragma unroll
  for (int off = 16; off > 0; off >>= 1) v += __shfl_xor(v, off, 32);
  return v;
}

// ---------------------------------------------------------------------------
// TDM: async DMA of a 16-row x 64-float fp32 tile (row stride 64 floats) from
// global memory directly into LDS. D# per CDNA5 ISA ch.8:
//   group0: [1:0]=count=1, [63:32]=lds_addr, [120:64]=global_addr, [127:126]=2
//   group1: [17:16]=data_size(4B), dim0=64, dim1=16, tile0=64, tile1=16,
//           dim0_stride=64; groups 2/3 zero (2-D tensor).
// Tracked on TENSORcnt; completion enforced with s_wait_tensorcnt.
// This toolchain exposes the 6-arg builtin (g0, g1, g2, g3, g4, cpol).
// ---------------------------------------------------------------------------
__device__ __forceinline__ void tdm_load_tile_16x64(const float* gsrc,
                                                    unsigned lds_byte_off) {
  const unsigned long long ga = (unsigned long long)(uintptr_t)gsrc;
  u32x4 g0;
  g0[0] = 1u;                                              // count=1, user mode
  g0[1] = lds_byte_off;                                    // lds_addr
  g0[2] = (unsigned)ga;                                    // global_addr[31:0]
  g0[3] = (unsigned)((ga >> 32) & 0x01FFFFFFu) | (2u << 30);  // addr[56:32] | type=2
  i32x8 g1;
  g1[0] = 2 << 16;    // data_size = 4 bytes
  g1[1] = 64 << 16;   // tensor_dim0 = 64        (bits 63:48)
  g1[2] = 16 << 16;   // tensor_dim1 = 16        (bits 95:80)
  g1[3] = 64 << 16;   // tile_dim0  = 64         (bits 127:112)
  g1[4] = 16;         // tile_dim1  = 16, tile_dim2 = 0
  g1[5] = 64;         // tensor_dim0_stride = 64 (bits 191:160)
  g1[6] = 0;
  g1[7] = 0;
  const i32x4 zz4 = {0, 0, 0, 0};
  const i32x8 zz8 = {0, 0, 0, 0, 0, 0, 0, 0};
  __builtin_amdgcn_tensor_load_to_lds(g0, g1, zz4, zz4, zz8, 0);
}

// ---------------------------------------------------------------------------
// Kernel 0: per-row prep. One wave32 per row (lane = dim k, Z == 32).
// ---------------------------------------------------------------------------
__global__ void tcvae_prep(const float* __restrict__ z, const float* __restrict__ mu,
                           const float* __restrict__ logvar,
                           float* __restrict__ Amat, float* __restrict__ Bmat,
                           float* __restrict__ half_t, float* __restrict__ Hs,
                           float* __restrict__ Ps, float* __restrict__ lqzx,
                           float* __restrict__ lpz, int N) {
  const int wave = threadIdx.x >> 5;
  const int k = threadIdx.x & 31;
  const int row = blockIdx.x * 8 + wave;
  if (row >= N) return;
  const size_t idx = (size_t)row * 32 + k;
  const float zv = z[idx], mv = mu[idx], lv = logvar[idx];
  const float ivv = __expf(-lv);
  const float he = -0.5f * (LOG2PI_F + lv);
  const float z2 = zv * zv;
  const size_t r64 = (size_t)row * 64;
  Amat[r64 + k] = z2;
  Amat[r64 + 32 + k] = zv;
  Bmat[r64 + k] = ivv;
  Bmat[r64 + 32 + k] = -2.0f * mv * ivv;
  half_t[idx] = he;
  const float d = zv - mv;
  const float e_lqzx = he - 0.5f * d * d * ivv;
  const float e_lpz = -0.5f * (LOG2PI_F + z2);
  const float e_P = mv * mv * ivv;
  const float sH = wave_sum(he);
  const float sP = wave_sum(e_P);
  const float sq = wave_sum(e_lqzx);
  const float sp = wave_sum(e_lpz);
  if (k == 0) { Hs[row] = sH; Ps[row] = sP; lqzx[row] = sq; lpz[row] = sp; }
}

// ---------------------------------------------------------------------------
// Kernel 1: fused WMMA GEMM (M=N=2048, K=64, fp32) + online LSE over j.
// One wave32 per 16-row i strip. B tiles are DMA'd into LDS by the Tensor
// Data Mover with double buffering: issue tile jt+1, s_wait_tensorcnt(1)
// (TDM ops complete in order per wave) then compute on tile jt.
// S[i,j] = logiw(i,j) + H[j] - 0.5*(Q[i,j] + P[j]),  Q = A.B^T
// ---------------------------------------------------------------------------
__global__ void __launch_bounds__(32)
tcvae_gemm_lse(const float* __restrict__ Amat, const float* __restrict__ Bmat,
               const float* __restrict__ Hs, const float* __restrict__ Ps,
               const int* __restrict__ dsz, float* __restrict__ lqz, int N) {
  __shared__ float Bt[2][16 * 64];
  const int lane = threadIdx.x;
  const int half_id = lane >> 4;
  const int l16 = lane & 15;
  const int i0 = blockIdx.x * 16;

  const int m = N - 1;
  const int nds = dsz[0];
  const float log_inv_m = -__logf((float)m);
  const float log_inv_n = -__logf((float)nds);
  const float log_strat = __logf((float)(nds - m) / ((float)nds * (float)m));

  const unsigned lds0 = (unsigned)(uintptr_t)&Bt[0][0];
  const unsigned lds1 = (unsigned)(uintptr_t)&Bt[1][0];

  // A fragments: 16 K-chunks of 4; lane layout per ISA (row = lane%16,
  // K pair = chunk*4 + (lane/16)*2).
  v2f a_frag[16];
#pragma unroll
  for (int kk = 0; kk < 16; ++kk) {
    const float* ap = Amat + (size_t)(i0 + l16) * 64 + kk * 4 + half_id * 2;
    a_frag[kk] = *(const v2f*)ap;
  }

  float runm[8], runs[8];
#pragma unroll
  for (int v = 0; v < 8; ++v) { runm[v] = -__builtin_inff(); runs[v] = 0.0f; }

  const int ntiles = N >> 4;
  // prologue: DMA tile 0 into buffer 0
  tdm_load_tile_16x64(Bmat, lds0);

  for (int jt = 0; jt < ntiles; ++jt) {
    if (jt + 1 < ntiles) {
      tdm_load_tile_16x64(Bmat + (size_t)(jt + 1) * 1024,
                          ((jt + 1) & 1) ? lds1 : lds0);
      __builtin_amdgcn_s_wait_tensorcnt(1);  // tile jt resident (in-order TDM)
    } else {
      __builtin_amdgcn_s_wait_tensorcnt(0);
    }
    asm volatile("" ::: "memory");
    const float* tile = &Bt[jt & 1][0];

    v8f c = {};
#pragma unroll
    for (int kk = 0; kk < 16; ++kk) {
      // B operand: col = lane%16 (j within tile), K pair = kk*4 + (lane/16)*2
      const v2f b = *(const v2f*)&tile[l16 * 64 + kk * 4 + half_id * 2];
      c = __builtin_amdgcn_wmma_f32_16x16x4_f32(false, a_frag[kk], false, b,
                                                (short)0, c, false, false);
    }

    const int j = jt * 16 + l16;
    const float Hj = Hs[j];
    const float Pj = Ps[j];
    const float liw_j = (j >= 2) ? log_inv_m : ((j == 1) ? log_strat : log_inv_n);
#pragma unroll
    for (int v = 0; v < 8; ++v) {
      const int i = i0 + half_id * 8 + v;               // C layout: M = v + 8*(lane/16)
      const float liw = ((j == 0) && (i == m - 1)) ? log_strat : liw_j;
      const float x = liw + Hj - 0.5f * (c[v] + Pj);
      const float nm = fmaxf(runm[v], x);
      runs[v] = runs[v] * __expf(runm[v] - nm) + __expf(x - nm);
      runm[v] = nm;
    }
  }

  // reduce LSE across the 16 lanes (N columns) of each half
#pragma unroll
  for (int v = 0; v < 8; ++v) {
    float rm = runm[v], rs = runs[v];
#pragma unroll
    for (int off = 1; off < 16; off <<= 1) {
      const float om = __shfl_xor(rm, off, 32);
      const float os = __shfl_xor(rs, off, 32);
      const float nm = fmaxf(rm, om);
      rs = rs * __expf(rm - nm) + os * __expf(om - nm);
      rm = nm;
    }
    if (l16 == 0) lqz[i0 + half_id * 8 + v] = rm + __logf(rs);
  }
}

// ---------------------------------------------------------------------------
// Kernel 2: per-dimension LSE over j (the elementwise-irreducible path).
// One wave32 per row i, lane = dim k; streams j with online LSE per lane,
// then sums the 32 per-dim LSEs across the wave -> logqz_prod_margs[i].
// ---------------------------------------------------------------------------
__global__ void tcvae_margs(const float* __restrict__ z, const float* __restrict__ mu,
                            const float* __restrict__ Bmat, const float* __restrict__ half_t,
                            const int* __restrict__ dsz,
                            float* __restrict__ lpm, int N) {
  const int wave = threadIdx.x >> 5;
  const int k = threadIdx.x & 31;
  const int i = blockIdx.x * 8 + wave;
  if (i >= N) return;

  const int m = N - 1;
  const int nds = dsz[0];
  const float log_inv_m = -__logf((float)m);
  const float log_inv_n = -__logf((float)nds);
  const float log_strat = __logf((float)(nds - m) / ((float)nds * (float)m));
  const float liw0 = (i == m - 1) ? log_strat : log_inv_n;

  const float zi = z[(size_t)i * 32 + k];
  float runm = -__builtin_inff(), runs = 0.0f;
  for (int j = 0; j < N; ++j) {
    __builtin_prefetch(mu + (size_t)(j + 16) * 32 + k, 0, 0);
    __builtin_prefetch(half_t + (size_t)(j + 16) * 32 + k, 0, 0);
    const float mv = mu[(size_t)j * 32 + k];
    const float ivv = Bmat[(size_t)j * 64 + k];
    const float he = half_t[(size_t)j * 32 + k];
    const float d = zi - mv;
    const float pre = he - 0.5f * d * d * ivv;
    const float liw = (j >= 2) ? log_inv_m : ((j == 1) ? log_strat : liw0);
    const float x = liw + pre;
    const float nm = fmaxf(runm, x);
    runs = runs * __expf(runm - nm) + __expf(x - nm);
    runm = nm;
  }
  const float lse = runm + __logf(runs);
  const float tot = wave_sum(lse);
  if (k == 0) lpm[i] = tot;
}

// ---------------------------------------------------------------------------
// Kernel 3: finalize loss + means.
// loss = mi*1 + tc*6 + dwkl*1 ; out = [loss(N), mi.mean, tc.mean, dwkl.mean]
// ---------------------------------------------------------------------------
__global__ void tcvae_finalize(const float* __restrict__ lqzx, const float* __restrict__ lqz,
                               const float* __restrict__ lpm, const float* __restrict__ lpz,
                               float* __restrict__ out, int N) {
  __shared__ float red[3][8];
  const int t = threadIdx.x;
  float smi = 0.0f, stc = 0.0f, sdw = 0.0f;
  for (int i = t; i < N; i += 256) {
    const float a = lqzx[i], b = lqz[i], c = lpm[i], d = lpz[i];
    const float mi = a - b, tc = b - c, dw = c - d;
    out[i] = mi * 1.0f + tc * 6.0f + dw * 1.0f;
    smi += mi; stc += tc; sdw += dw;
  }
  smi = wave_sum(smi); stc = wave_sum(stc); sdw = wave_sum(sdw);
  const int w = t >> 5, l = t & 31;
  if (l == 0) { red[0][w] = smi; red[1][w] = stc; red[2][w] = sdw; }
  __syncthreads();
  if (t == 0) {
    float a = 0.0f, b = 0.0f, c = 0.0f;
    for (int ww = 0; ww < 8; ++ww) { a += red[0][ww]; b += red[1][ww]; c += red[2][ww]; }
    const float invN = 1.0f / (float)N;
    out[N] = a * invN; out[N + 1] = b * invN; out[N + 2] = c * invN;
  }
}

extern "C" void kernel_launch(void* const* d_in, const int* in_sizes, int n_in,
                              void* d_out, int out_size, void* d_ws, size_t ws_size,
                              hipStream_t stream) {
  const float* z = (const float*)d_in[0];
  const float* mu = (const float*)d_in[1];
  const float* logvar = (const float*)d_in[2];
  const int* dsz = (const int*)d_in[3];
  const int N = in_sizes[0] / 32;  // 2048, Z fixed at 32 (one wave32 lane per dim)

  float* A      = (float*)d_ws;               // N*64
  float* B      = A + (size_t)N * 64;         // N*64
  float* half_t = B + (size_t)N * 64;         // N*32
  float* Hs     = half_t + (size_t)N * 32;    // N
  float* Ps     = Hs + N;                     // N
  float* lqzx   = Ps + N;                     // N
  float* lpz    = lqzx + N;                   // N
  float* lqz    = lpz + N;                    // N
  float* lpm    = lqz + N;                    // N
  float* out    = (float*)d_out;

  tcvae_prep<<<N / 8, 256, 0, stream>>>(z, mu, logvar, A, B, half_t, Hs, Ps, lqzx, lpz, N);
  tcvae_gemm_lse<<<N / 16, 32, 0, stream>>>(A, B, Hs, Ps, dsz, lqz, N);
  tcvae_margs<<<N / 8, 256, 0, stream>>>(z, mu, B, half_t, dsz, lpm, N);
  tcvae_finalize<<<1, 256, 0, stream>>>(lqzx, lqz, lpm, lpz, out, N);
}